// TensorProductLayer_36017595744835
// MI455X (gfx1250) — compile-verified
//
#include <hip/hip_runtime.h>
#include <cstdint>

// ---------------------------------------------------------------------------
// out[i,l] = sum_{j,k} p3[i,j,l] * p5[i,j,k,l] * p3[i,k,l]
// N=100000, J=3, L=128, fp32.  ~666 MB traffic, ~0.5 FLOP/byte -> pure HBM
// streaming problem.  Strategy: async global->LDS (ASYNCcnt) double-buffered
// pipeline, B128 everywhere, wave32-sized blocks, NT stores.
// ---------------------------------------------------------------------------

#define THREADS          128                     // 4 waves (wave32)
#define TILE_ATOMS       4                       // atoms per pipeline stage
#define STAGES           8                       // stages per block
#define ATOMS_PER_BLOCK  (TILE_ATOMS * STAGES)   // 32

#define P3_F4_ATOM   96                          // 3*128/4 float4 per atom
#define P5_F4_ATOM   288                         // 9*128/4 float4 per atom
#define P3_F4_STAGE  (TILE_ATOMS * P3_F4_ATOM)   // 384 float4 (6 KB)
#define P5_F4_STAGE  (TILE_ATOMS * P5_F4_ATOM)   // 1152 float4 (18 KB)
#define P3_COPIES    (P3_F4_STAGE / THREADS)     // 3 async b128 per thread
#define P5_COPIES    (P5_F4_STAGE / THREADS)     // 9 async b128 per thread
// per-wave ASYNCcnt increment per stage = P3_COPIES + P5_COPIES = 12

// Native clang vector type for the nontemporal B128 store.
typedef float v4f __attribute__((ext_vector_type(4)));

// Low 32 bits of a flat pointer into LDS == wave-relative LDS byte offset
// (flat LDS aperture keeps the offset in addr[31:0]).
__device__ __forceinline__ unsigned lds_addr32(const void* p) {
  return (unsigned)(uintptr_t)p;
}

// One ASYNCcnt-tracked 16-byte global->LDS copy (GVS addressing:
// SADDR = uniform array base, VADDR = per-lane 32-bit byte offset).
__device__ __forceinline__ void async_b128(unsigned lds, unsigned gbyte,
                                           const void* sbase) {
  asm volatile("global_load_async_to_lds_b128 %0, %1, %2"
               :
               : "v"(lds), "v"(gbyte), "s"(sbase)
               : "memory");
}

__global__ __launch_bounds__(THREADS)
void tpq_quadform_kernel(const float4* __restrict__ p3,
                         const float4* __restrict__ p5,
                         v4f* __restrict__ out,
                         int n) {
  __shared__ float4 sm_p3[2][P3_F4_STAGE];   // 12 KB
  __shared__ float4 sm_p5[2][P5_F4_STAGE];   // 36 KB  -> 48 KB total

  const int t = threadIdx.x;
  const long long a_base = (long long)blockIdx.x * ATOMS_PER_BLOCK;

  // Clamp sources to last valid float4 (tail stays in-bounds; compute is
  // guarded by atom < n).
  const unsigned p3_max = (unsigned)n * P3_F4_ATOM - 1u;
  const unsigned p5_max = (unsigned)n * P5_F4_ATOM - 1u;

  unsigned lds_p3[2], lds_p5[2];
  lds_p3[0] = lds_addr32(&sm_p3[0][0]);
  lds_p3[1] = lds_addr32(&sm_p3[1][0]);
  lds_p5[0] = lds_addr32(&sm_p5[0][0]);
  lds_p5[1] = lds_addr32(&sm_p5[1][0]);

  auto issue = [&](int s) {
    const int buf = s & 1;
    const unsigned a0 = (unsigned)(a_base + (long long)s * TILE_ATOMS);
#pragma unroll
    for (int k = 0; k < P3_COPIES; ++k) {
      const unsigned local = (unsigned)(t + k * THREADS);
      unsigned g = a0 * P3_F4_ATOM + local;
      g = g > p3_max ? p3_max : g;
      async_b128(lds_p3[buf] + local * 16u, g * 16u, p3);
    }
#pragma unroll
    for (int k = 0; k < P5_COPIES; ++k) {
      const unsigned local = (unsigned)(t + k * THREADS);
      unsigned g = a0 * P5_F4_ATOM + local;
      g = g > p5_max ? p5_max : g;
      async_b128(lds_p5[buf] + local * 16u, g * 16u, p5);
    }
  };

  issue(0);  // prologue: stage 0 in flight

  const int atom_local = t >> 5;   // 0..3 : atom within stage
  const int grp        = t & 31;   // 0..31: float4 group along L

  for (int s = 0; s < STAGES; ++s) {
    const int buf = s & 1;
    if (s + 1 < STAGES) {
      issue(s + 1);
      // Async loads complete in order: <=12 outstanding => stage s landed.
      asm volatile("s_wait_asynccnt 12" ::: "memory");
    } else {
      asm volatile("s_wait_asynccnt 0" ::: "memory");
    }
    __syncthreads();  // all waves' stage-s data visible before any read

    const long long atom = a_base + (long long)s * TILE_ATOMS + atom_local;
    if (atom < n) {
      float4 av[3];
#pragma unroll
      for (int j = 0; j < 3; ++j)
        av[j] = sm_p3[buf][(atom_local * 3 + j) * 32 + grp];

      v4f acc = (v4f)(0.0f);
#pragma unroll
      for (int j = 0; j < 3; ++j) {
        v4f q = (v4f)(0.0f);
#pragma unroll
        for (int k = 0; k < 3; ++k) {
          const float4 m = sm_p5[buf][(atom_local * 9 + j * 3 + k) * 32 + grp];
          q.x = fmaf(m.x, av[k].x, q.x);
          q.y = fmaf(m.y, av[k].y, q.y);
          q.z = fmaf(m.z, av[k].z, q.z);
          q.w = fmaf(m.w, av[k].w, q.w);
        }
        acc.x = fmaf(av[j].x, q.x, acc.x);
        acc.y = fmaf(av[j].y, q.y, acc.y);
        acc.z = fmaf(av[j].z, q.z, acc.z);
        acc.w = fmaf(av[j].w, q.w, acc.w);
      }
      // Streaming output (666 MB working set >> 192 MB L2): NT store.
      __builtin_nontemporal_store(acc, &out[(size_t)atom * 32 + grp]);
    }
    __syncthreads();  // buffer buf free for reuse by stage s+2
  }
}

extern "C" void kernel_launch(void* const* d_in, const int* in_sizes, int n_in,
                              void* d_out, int out_size, void* d_ws, size_t ws_size,
                              hipStream_t stream) {
  const float4* p3 = (const float4*)d_in[0];   // [N,3,128] f32
  const float4* p5 = (const float4*)d_in[1];   // [N,3,3,128] f32
  v4f* out = (v4f*)d_out;                      // [N,128] f32

  const int n = in_sizes[0] / 384;             // N = |p3| / (3*128)
  const int blocks = (n + ATOMS_PER_BLOCK - 1) / ATOMS_PER_BLOCK;

  tpq_quadform_kernel<<<blocks, THREADS, 0, stream>>>(p3, p5, out, n);
}